// GAT_70720931496420
// MI455X (gfx1250) — compile-verified
//
#include <hip/hip_runtime.h>
#include <hip/hip_bf16.h>
#include <math.h>

// ---------------------------------------------------------------------------
// GAT forward (2 GAT layers + 2-layer MLP), MI455X / gfx1250.
// GEMMs use V_WMMA_F32_16X16X4_F32 (fp32 matrix cores, wave32).
// Edge phase is HBM-bandwidth bound (~2 GB traffic -> ~90us @ 23.3 TB/s);
// organized so the channel dim is the fast (coalesced) dimension.
// ---------------------------------------------------------------------------

#define N_NODES 100000
#define N_EDGES 1600000
#define IN_DIM  128
#define HID     64

typedef __attribute__((ext_vector_type(2))) float v2f;
typedef __attribute__((ext_vector_type(8))) float v8f;

// ---------------------------------------------------------------------------
// WMMA f32 GEMM: out[M x 64] = (relu_in ? relu(A) : A)[M x K] * W[K x 64] + bias
// One 16x16 output tile per wave; 8 waves per 256-thread block -> 32x64/block.
// VGPR layouts per CDNA5 ISA 7.12.2:
//   A 16x4  : lanes 0-15 hold M=lane, K={k,k+1}; lanes 16-31 hold M=lane-16,
//             K={k+2,k+3} (2 VGPRs / v2f per lane).
//   B 4x16  : VGPR v, lanes 0-15 -> row K=k+v, lanes 16-31 -> row K=k+v+2.
//   C 16x16 : VGPR r, lanes 0-15 -> (M=r, N=lane), lanes 16-31 -> (M=r+8).
// All control flow around the WMMA is wave-uniform (EXEC stays all ones).
// ---------------------------------------------------------------------------
__global__ __launch_bounds__(256) void gemm_wmma_f32(
    const float* __restrict__ A, const float* __restrict__ W,
    const float* __restrict__ bias, float* __restrict__ out,
    int M, int K, int relu_in)
{
    const int lane = threadIdx.x & 31;
    const int wave = threadIdx.x >> 5;       // 0..7
    const int row0 = blockIdx.x * 32 + (wave >> 2) * 16;
    const int col0 = (wave & 3) * 16;        // 64 output cols = 4 col tiles
    if (row0 >= M) return;                   // wave-uniform

    const int half = lane >> 4;              // 0 -> K {0,1}; 1 -> K {2,3}
    const int l16  = lane & 15;

    const float* __restrict__ arow = A + (size_t)(row0 + l16) * K;

    v8f acc = {};
    for (int k = 0; k < K; k += 4) {
        const int ka = k + half * 2;
        float a0 = arow[ka];
        float a1 = arow[ka + 1];
        if (relu_in) { a0 = fmaxf(a0, 0.0f); a1 = fmaxf(a1, 0.0f); }
        v2f av; av[0] = a0; av[1] = a1;

        v2f bv;
        bv[0] = W[(size_t)(ka)     * HID + col0 + l16];
        bv[1] = W[(size_t)(ka + 1) * HID + col0 + l16];

        acc = __builtin_amdgcn_wmma_f32_16x16x4_f32(
            /*neg_a=*/false, av, /*neg_b=*/false, bv,
            /*c_mod=*/(short)0, acc, /*reuse_a=*/false, /*reuse_b=*/false);
    }

    const int col = col0 + l16;
    const float b = bias ? bias[col] : 0.0f;
#pragma unroll
    for (int r = 0; r < 8; ++r) {
        const int row = row0 + r + half * 8;
        out[(size_t)row * HID + col] = acc[r] + b;
    }
}

// ---------------------------------------------------------------------------
// Per-node attention dots: aL[n] = <xh[n], att_l>, aR[n] = <xh[n], att_r>.
// One wave32 per node, 2 channels/lane, shuffle reduction.
// ---------------------------------------------------------------------------
__global__ __launch_bounds__(256) void node_alpha(
    const float* __restrict__ xh, const float* __restrict__ al,
    const float* __restrict__ ar, float* __restrict__ aL,
    float* __restrict__ aR, int n)
{
    const int node = (int)((blockIdx.x * blockDim.x + threadIdx.x) >> 5);
    const int lane = threadIdx.x & 31;
    if (node >= n) return;
    const float* __restrict__ row = xh + (size_t)node * HID;
    float l = 0.0f, r = 0.0f;
#pragma unroll
    for (int c = lane; c < HID; c += 32) {
        const float v = row[c];
        l += v * al[c];
        r += v * ar[c];
    }
#pragma unroll
    for (int off = 16; off > 0; off >>= 1) {
        l += __shfl_down(l, off, 32);
        r += __shfl_down(r, off, 32);
    }
    if (lane == 0) { aL[node] = l; aR[node] = r; }
}

// ---------------------------------------------------------------------------
// Init per-layer state: amax=-inf, denom=0, agg[N*64]=0.
// ---------------------------------------------------------------------------
__global__ __launch_bounds__(256) void init_state(
    float* __restrict__ amax, float* __restrict__ denom,
    float* __restrict__ agg, int n)
{
    const int i = blockIdx.x * blockDim.x + threadIdx.x;
    if (i < n) { amax[i] = -INFINITY; denom[i] = 0.0f; }
    if (i < n * HID) agg[i] = 0.0f;
}

// Float atomic max via signed/unsigned int ordering (valid with -inf init).
__device__ __forceinline__ void atomicMaxF(float* addr, float val) {
    if (val >= 0.0f)
        atomicMax((int*)addr, __float_as_int(val));
    else
        atomicMin((unsigned int*)addr, __float_as_uint(val));
}

// ---------------------------------------------------------------------------
// Edge pass 1: a = leaky_relu(aL[src] + aR[dst]); edge_a = a; amax[dst] max=.
// ---------------------------------------------------------------------------
__global__ __launch_bounds__(256) void edge_logit_max(
    const long long* __restrict__ src, const long long* __restrict__ dst,
    const float* __restrict__ aL, const float* __restrict__ aR,
    float* __restrict__ edge_a, float* __restrict__ amax, int E)
{
    const int e = blockIdx.x * blockDim.x + threadIdx.x;
    if (e >= E) return;
    const int s = (int)src[e];
    const int d = (int)dst[e];
    float a = aL[s] + aR[d];
    a = (a >= 0.0f) ? a : 0.2f * a;
    edge_a[e] = a;
    atomicMaxF(&amax[d], a);
}

// ---------------------------------------------------------------------------
// Edge pass 2: e = exp(a - amax[dst]); edge_a = e; denom[dst] += e.
// ---------------------------------------------------------------------------
__global__ __launch_bounds__(256) void edge_exp_sum(
    const long long* __restrict__ dst, float* __restrict__ edge_a,
    const float* __restrict__ amax, float* __restrict__ denom, int E)
{
    const int e = blockIdx.x * blockDim.x + threadIdx.x;
    if (e >= E) return;
    const int d = (int)dst[e];
    const float v = expf(edge_a[e] - amax[d]);
    edge_a[e] = v;
    atomicAdd(&denom[d], v);
}

// ---------------------------------------------------------------------------
// Edge pass 3: agg[dst] += xh[src] * (e / (denom[dst] + 1e-16)).
// One thread per (edge, channel); channel is the lane-fast dim so reads of
// xh[src*64..] and atomics on agg[dst*64..] are contiguous 64-float rows.
// ---------------------------------------------------------------------------
__global__ __launch_bounds__(256) void edge_scatter(
    const long long* __restrict__ src, const long long* __restrict__ dst,
    const float* __restrict__ edge_a, const float* __restrict__ denom,
    const float* __restrict__ xh, float* __restrict__ agg, int E)
{
    const int t = blockIdx.x * blockDim.x + threadIdx.x;   // E*64 < 2^31
    const int e = t >> 6;
    const int c = t & 63;
    if (e >= E) return;
    const int s = (int)src[e];
    const int d = (int)dst[e];
    const float alpha = edge_a[e] / (denom[d] + 1e-16f);
    atomicAdd(&agg[(size_t)d * HID + c], xh[(size_t)s * HID + c] * alpha);
}

// ---------------------------------------------------------------------------
// Host-side launch sequence.
// ---------------------------------------------------------------------------
extern "C" void kernel_launch(void* const* d_in, const int* in_sizes, int n_in,
                              void* d_out, int out_size, void* d_ws, size_t ws_size,
                              hipStream_t stream) {
    (void)in_sizes; (void)n_in; (void)out_size; (void)ws_size;

    const float*     x    = (const float*)d_in[0];
    const long long* ei   = (const long long*)d_in[1];   // int64 [2, E]
    const float*     W0   = (const float*)d_in[2];
    const float*     al0  = (const float*)d_in[3];
    const float*     ar0  = (const float*)d_in[4];
    const float*     W1   = (const float*)d_in[5];
    const float*     al1  = (const float*)d_in[6];
    const float*     ar1  = (const float*)d_in[7];
    const float*     Wp1  = (const float*)d_in[8];
    const float*     bp1  = (const float*)d_in[9];
    const float*     Wp2  = (const float*)d_in[10];
    const float*     bp2  = (const float*)d_in[11];
    float* out = (float*)d_out;

    const long long* src = ei;
    const long long* dst = ei + N_EDGES;

    // Workspace layout (floats): ~59 MB total.
    float* ws     = (float*)d_ws;
    float* bufA   = ws;                               // N*64 xh buffer
    float* bufB   = bufA + (size_t)N_NODES * HID;     // N*64 aggregate buffer
    float* aL     = bufB + (size_t)N_NODES * HID;     // N
    float* aR     = aL + N_NODES;                     // N
    float* amax   = aR + N_NODES;                     // N
    float* denom  = amax + N_NODES;                   // N
    float* edgeA  = denom + N_NODES;                  // E

    const int gemmGrid    = (N_NODES + 31) / 32;                  // 3125
    const int alphaGrid   = (N_NODES * 32 + 255) / 256;           // waves: 1/node
    const int initGrid    = (N_NODES * HID + 255) / 256;
    const int edgeGrid    = (N_EDGES + 255) / 256;
    const int scatterGrid = (N_EDGES * 64) / 256;                 // exact

    // ---- GAT layer 0: xh0 = x @ W0 -----------------------------------------
    gemm_wmma_f32<<<gemmGrid, 256, 0, stream>>>(x, W0, nullptr, bufA,
                                                N_NODES, IN_DIM, 0);
    node_alpha<<<alphaGrid, 256, 0, stream>>>(bufA, al0, ar0, aL, aR, N_NODES);
    init_state<<<initGrid, 256, 0, stream>>>(amax, denom, bufB, N_NODES);
    edge_logit_max<<<edgeGrid, 256, 0, stream>>>(src, dst, aL, aR, edgeA, amax, N_EDGES);
    edge_exp_sum<<<edgeGrid, 256, 0, stream>>>(dst, edgeA, amax, denom, N_EDGES);
    edge_scatter<<<scatterGrid, 256, 0, stream>>>(src, dst, edgeA, denom,
                                                  bufA, bufB, N_EDGES);
    // bufB = layer-0 output (pre-ReLU); ReLU fused into next GEMM's loads.

    // ---- GAT layer 1: xh1 = relu(h0) @ W1 ----------------------------------
    gemm_wmma_f32<<<gemmGrid, 256, 0, stream>>>(bufB, W1, nullptr, bufA,
                                                N_NODES, HID, 1);
    node_alpha<<<alphaGrid, 256, 0, stream>>>(bufA, al1, ar1, aL, aR, N_NODES);
    init_state<<<initGrid, 256, 0, stream>>>(amax, denom, bufB, N_NODES);
    edge_logit_max<<<edgeGrid, 256, 0, stream>>>(src, dst, aL, aR, edgeA, amax, N_EDGES);
    edge_exp_sum<<<edgeGrid, 256, 0, stream>>>(dst, edgeA, amax, denom, N_EDGES);
    edge_scatter<<<scatterGrid, 256, 0, stream>>>(src, dst, edgeA, denom,
                                                  bufA, bufB, N_EDGES);
    // bufB = layer-1 output (pre-ReLU).

    // ---- post_mp: relu(h1) @ Wp1 + bp1, then @ Wp2 + bp2 -------------------
    gemm_wmma_f32<<<gemmGrid, 256, 0, stream>>>(bufB, Wp1, bp1, bufA,
                                                N_NODES, HID, 1);
    gemm_wmma_f32<<<gemmGrid, 256, 0, stream>>>(bufA, Wp2, bp2, out,
                                                N_NODES, HID, 0);
}